// VLADPooling_46016279609383
// MI455X (gfx1250) — compile-verified
//
#include <hip/hip_runtime.h>
#include <math.h>

typedef float v2f __attribute__((ext_vector_type(2)));
typedef float v8f __attribute__((ext_vector_type(8)));
typedef int   v4i __attribute__((ext_vector_type(4)));
typedef __attribute__((address_space(1))) v4i v4i_g;   // global int4
typedef __attribute__((address_space(3))) v4i v4i_l;   // LDS int4

#define B_DIM 32
#define N_PTS 4096
#define D_DIM 128
#define K_CL  64
#define ALPHA_C 100.0f
#define ROWS 64
#define BLOCKS_PER_B 16
#define ITERS (N_PTS / (ROWS * BLOCKS_PER_B))   // 4
#define THREADS 256

// Padded LDS strides (floats) to avoid 64-bank conflicts on column accesses
#define XS_S 132   // rows of x chunk     (132 % 64 == 4, 16B-aligned rows)
#define CS_S 132   // rows of centroids
#define XC_S 65    // xc rows             (odd -> conflict-free row walk)
#define AT_S 68    // soft-assign^T rows

// Double-buffered x chunks for async global->LDS staging.
#define XS0_OFF 0
#define XS1_OFF (XS0_OFF + ROWS * XS_S)
#define CS_OFF  (XS1_OFF + ROWS * XS_S)
#define XC_OFF  (CS_OFF + K_CL * CS_S)
#define AT_OFF  (XC_OFF + ROWS * XC_S)
#define X2_OFF  (AT_OFF + K_CL * AT_S)
#define C2_OFF  (X2_OFF + ROWS)
#define SMEM_FLOATS (C2_OFF + K_CL)
#define SMEM_BYTES (SMEM_FLOATS * 4)             // ~133 KB < 320 KB WGP LDS

#define WMMA_F32(A, B, C) \
  __builtin_amdgcn_wmma_f32_16x16x4_f32(false, (A), false, (B), (short)0, (C), false, false)

#if __has_builtin(__builtin_amdgcn_global_load_async_to_lds_b128)
#define ASYNC_STAGE 1
#else
#define ASYNC_STAGE 0
#endif

#if __has_builtin(__builtin_amdgcn_s_wait_asynccnt)
#define WAIT_ASYNC0() __builtin_amdgcn_s_wait_asynccnt(0)
#else
#define WAIT_ASYNC0() asm volatile("s_wait_asynccnt 0x0" ::: "memory")
#endif

// Stage one 64x128 f32 chunk into padded LDS. Each thread moves 8x 16B.
__device__ __forceinline__ void stage_chunk(float* __restrict__ dst,
                                            const float* __restrict__ gsrc,
                                            int tid) {
  for (int f = tid; f < (ROWS * D_DIM) / 4; f += THREADS) {
    int r = f >> 5, c = f & 31;
#if ASYNC_STAGE
    __builtin_amdgcn_global_load_async_to_lds_b128(
        (v4i_g*)(gsrc + (size_t)f * 4),
        (v4i_l*)(dst + r * XS_S + c * 4), 0, 0);
#else
    ((float4*)(dst + r * XS_S))[c] = ((const float4*)gsrc)[f];
#endif
  }
}

__global__ void vlad_zero(float* __restrict__ out_acc, float* __restrict__ asum_g) {
  int i = blockIdx.x * blockDim.x + threadIdx.x;
  if (i < B_DIM * K_CL * D_DIM) out_acc[i] = 0.0f;
  if (i < B_DIM * K_CL) asum_g[i] = 0.0f;
}

__global__ void __launch_bounds__(THREADS)
vlad_main(const float* __restrict__ x, const float* __restrict__ cent,
          float* __restrict__ out_acc, float* __restrict__ asum_g) {
  extern __shared__ float sm[];
  float* cs = sm + CS_OFF;
  float* xc = sm + XC_OFF;
  float* at = sm + AT_OFF;
  float* x2 = sm + X2_OFF;
  float* c2 = sm + C2_OFF;

  const int tid  = threadIdx.x;
  const int lane = tid & 31;
  const int wave = tid >> 5;
  const int b    = blockIdx.y;
  const int blk  = blockIdx.x;
  const int lr   = lane & 15;   // fragment row/col within 16
  const int lh   = lane >> 4;   // half-wave select

  const float* xbase = x + (size_t)b * N_PTS * D_DIM
                         + (size_t)blk * ITERS * ROWS * D_DIM;

#if ASYNC_STAGE
  // Kick off chunk 0 staging immediately (ASYNCcnt-tracked, overlaps with
  // the centroid staging below).
  stage_chunk(sm + XS0_OFF, xbase, tid);
#endif

  // Stage centroids (64x128) into LDS with padded stride, compute c2.
  {
    const float4* src = (const float4*)cent;
    for (int f = tid; f < (K_CL * D_DIM) / 4; f += THREADS) {
      int r = f >> 5, c = f & 31;
      ((float4*)(cs + r * CS_S))[c] = src[f];
    }
  }
  __syncthreads();
  if (tid < K_CL) {
    float s = 0.0f;
    for (int d = 0; d < D_DIM; ++d) { float v = cs[tid * CS_S + d]; s += v * v; }
    c2[tid] = s;
  }

  v8f cacc0 = {}, cacc1 = {}, cacc2 = {}, cacc3 = {};
  float asum_reg = 0.0f;

  // Per-wave tile bases: all of a wave's tiles share the same M-tile so the
  // A fragment is loaded once per k-step and reused.
  const int mB = (wave >> 1) << 4;          // M tile (rows / clusters)
  const int nB = (wave & 1) << 5;           // phase-B N base: 0 or 32
  const int nD = (wave & 1) << 6;           // phase-D N base: 0 or 64

  for (int it = 0; it < ITERS; ++it) {
    float* xs = sm + ((it & 1) ? XS1_OFF : XS0_OFF);
#if ASYNC_STAGE
    WAIT_ASYNC0();            // my async stores into xs (chunk `it`) landed
    __syncthreads();          // everyone's landed; prev readers of `next` done
    if (it + 1 < ITERS) {     // overlap next chunk's staging with compute
      float* xn = sm + ((it & 1) ? XS0_OFF : XS1_OFF);
      stage_chunk(xn, xbase + (size_t)(it + 1) * ROWS * D_DIM, tid);
    }
#else
    __syncthreads();
    stage_chunk(xs, xbase + (size_t)it * ROWS * D_DIM, tid);
    __syncthreads();
#endif
    if (tid < ROWS) {
      float s = 0.0f;
      for (int d = 0; d < D_DIM; ++d) { float v = xs[tid * XS_S + d]; s += v * v; }
      x2[tid] = s;
    }

    // ---- Phase B: XC[64x64] = Xchunk * C^T via V_WMMA_F32_16X16X4_F32 ----
    {
      v8f acc0 = {}, acc1 = {};
      #pragma unroll
      for (int k0 = 0; k0 < D_DIM; k0 += 4) {
        const int kk = k0 + 2 * lh;
        v2f a  = *(const v2f*)&xs[(mB + lr) * XS_S + kk];
        v2f b0 = *(const v2f*)&cs[(nB + lr) * CS_S + kk];
        v2f b1 = *(const v2f*)&cs[(nB + 16 + lr) * CS_S + kk];
        acc0 = WMMA_F32(a, b0, acc0);
        acc1 = WMMA_F32(a, b1, acc1);
      }
      #pragma unroll
      for (int v = 0; v < 8; ++v) {
        xc[(mB + v + 8 * lh) * XC_S + (nB + lr)]      = acc0[v];
        xc[(mB + v + 8 * lh) * XC_S + (nB + 16 + lr)] = acc1[v];
      }
    }
    __syncthreads();

    // ---- Phase C: per-row softmax over 64 clusters (threads 0..63) ----
    if (tid < ROWS) {
      const float xv = x2[tid];
      float mind = 1e30f;
      for (int k = 0; k < K_CL; ++k) {
        float d2   = xv + c2[k] - 2.0f * xc[tid * XC_S + k];
        float dist = sqrtf(fmaxf(d2, 0.0f));
        xc[tid * XC_S + k] = dist;
        mind = fminf(mind, dist);
      }
      float sum = 0.0f;
      for (int k = 0; k < K_CL; ++k) {
        float e = __expf(ALPHA_C * (mind - xc[tid * XC_S + k]));
        xc[tid * XC_S + k] = e;
        sum += e;
      }
      const float inv = 1.0f / sum;
      for (int k = 0; k < K_CL; ++k)
        at[k * AT_S + tid] = xc[tid * XC_S + k] * inv;   // transposed store
    }
    __syncthreads();
    if (tid < K_CL) {
      float s = 0.0f;
      for (int r = 0; r < ROWS; ++r) s += at[tid * AT_S + r];
      asum_reg += s;
    }

    // ---- Phase D: ACC[64x128] += A^T(64x64) * Xchunk(64x128) ----
    #pragma unroll
    for (int k0 = 0; k0 < ROWS; k0 += 4) {
      const int kk = k0 + 2 * lh;
      v2f a = *(const v2f*)&at[(mB + lr) * AT_S + kk];
      v2f b0, b1, b2, b3;
      b0.x = xs[kk * XS_S + nD + lr];       b0.y = xs[(kk + 1) * XS_S + nD + lr];
      b1.x = xs[kk * XS_S + nD + 16 + lr];  b1.y = xs[(kk + 1) * XS_S + nD + 16 + lr];
      b2.x = xs[kk * XS_S + nD + 32 + lr];  b2.y = xs[(kk + 1) * XS_S + nD + 32 + lr];
      b3.x = xs[kk * XS_S + nD + 48 + lr];  b3.y = xs[(kk + 1) * XS_S + nD + 48 + lr];
      cacc0 = WMMA_F32(a, b0, cacc0);
      cacc1 = WMMA_F32(a, b1, cacc1);
      cacc2 = WMMA_F32(a, b2, cacc2);
      cacc3 = WMMA_F32(a, b3, cacc3);
    }
  }

  // Flush partial [K,D] accumulators with fp32 atomics.
  #pragma unroll
  for (int j = 0; j < 4; ++j) {
    v8f acc = (j == 0) ? cacc0 : (j == 1) ? cacc1 : (j == 2) ? cacc2 : cacc3;
    const int n0 = nD + (j << 4);
    #pragma unroll
    for (int v = 0; v < 8; ++v) {
      const int k = mB + v + 8 * lh;
      const int d = n0 + lr;
      atomicAdd(&out_acc[((size_t)b * K_CL + k) * D_DIM + d], acc[v]);
    }
  }
  if (tid < K_CL) atomicAdd(&asum_g[b * K_CL + tid], asum_reg);
}

__global__ void vlad_finalize(float* __restrict__ out,
                              const float* __restrict__ asum_g,
                              const float* __restrict__ cent) {
  __shared__ float red[THREADS];
  const int b = blockIdx.x;
  const int tid = threadIdx.x;
  float ss = 0.0f;
  for (int i = tid; i < K_CL * D_DIM; i += THREADS) {
    const int k = i >> 7;
    const int d = i & 127;
    float v = out[(size_t)b * K_CL * D_DIM + i] - asum_g[b * K_CL + k] * cent[k * D_DIM + d];
    out[(size_t)b * K_CL * D_DIM + i] = v;
    ss += v * v;
  }
  red[tid] = ss;
  __syncthreads();
  for (int s = THREADS / 2; s > 0; s >>= 1) {
    if (tid < s) red[tid] += red[tid + s];
    __syncthreads();
  }
  const float inv = 1.0f / fmaxf(sqrtf(red[0]), 1e-12f);
  for (int i = tid; i < K_CL * D_DIM; i += THREADS)
    out[(size_t)b * K_CL * D_DIM + i] *= inv;
}

extern "C" void kernel_launch(void* const* d_in, const int* in_sizes, int n_in,
                              void* d_out, int out_size, void* d_ws, size_t ws_size,
                              hipStream_t stream) {
  (void)in_sizes; (void)n_in; (void)out_size; (void)ws_size;
  const float* x    = (const float*)d_in[0];
  const float* cent = (const float*)d_in[1];
  float* out    = (float*)d_out;
  float* asum_g = (float*)d_ws;   // 32*64 floats = 8 KB scratch

  const int ztotal = B_DIM * K_CL * D_DIM;
  vlad_zero<<<dim3((ztotal + THREADS - 1) / THREADS), THREADS, 0, stream>>>(out, asum_g);

  dim3 grid(BLOCKS_PER_B, B_DIM);
  vlad_main<<<grid, THREADS, SMEM_BYTES, stream>>>(x, cent, out, asum_g);

  vlad_finalize<<<B_DIM, THREADS, 0, stream>>>(out, asum_g, cent);
}